// EDQPTHSolver_45363444580883
// MI455X (gfx1250) — compile-verified
//
#include <hip/hip_runtime.h>

// ---------------------------------------------------------------------------
// Batched PDIPM QP solver for MI455X (gfx1250).
//   min 0.5 x^T Q x + q^T x  s.t. Gx <= h, Ax = b
// B=128, NZ=160, NINEQ=240, NEQ=40, 15 iterations, fp32 throughout.
// - Matrix work: V_WMMA_F32_16X16X4_F32 (fp32 accuracy for Cholesky KKT solve)
// - G staged into LDS via Tensor Data Mover (tensor_load_to_lds + TENSORcnt)
// - Whole working set (<2MB) is L2-resident; problem is compute/latency bound.
// ---------------------------------------------------------------------------

#define BATCH 128
#define PNZ 160
#define PNINEQ 240
#define PNEQ 40
#define NITER 15
#define PSIGMA 0.1f
#define ATPAD 48  // A^T padded column stride (full 16-wide tiles, no load guards)

typedef float v2f __attribute__((ext_vector_type(2)));
typedef float v8f __attribute__((ext_vector_type(8)));
typedef unsigned int v4u __attribute__((ext_vector_type(4)));
typedef int v4i __attribute__((ext_vector_type(4)));
typedef int v8i __attribute__((ext_vector_type(8)));

__device__ __forceinline__ v8f wmma4(v2f a, v2f b, v8f c) {
  // D = A(16x4 f32) * B(4x16 f32) + C(16x16 f32)
  return __builtin_amdgcn_wmma_f32_16x16x4_f32(false, a, false, b,
                                               (short)0, c, false, false);
}

// TDM: DMA a 2-D fp32 tile (dim1 rows x dim0 cols, row stride stride0 elems)
// from global memory into LDS at byte offset lds_byte. D# per ISA 08 §8.3/8.4.
__device__ __forceinline__ void tdm_load_2d_f32(const void* gptr,
                                                unsigned lds_byte,
                                                unsigned dim0, unsigned dim1,
                                                unsigned stride0) {
  unsigned long long ga = (unsigned long long)(uintptr_t)gptr;
  v4u g0;
  g0.x = 1u;                                     // count=1, user descriptor
  g0.y = lds_byte;                               // lds_addr [63:32]
  g0.z = (unsigned)(ga & 0xffffffffu);           // global_addr [95:64]
  g0.w = (unsigned)((ga >> 32) & 0x1ffffffu)     // global_addr [120:96]
         | (2u << 30);                           // type=2 ("image")
  v8i g1;
  g1[0] = (int)(2u << 16);                       // data_size=2 -> 4 bytes
  g1[1] = (int)((dim0 & 0xffffu) << 16);         // tensor_dim0[15:0] @ bits 63:48
  g1[2] = (int)((dim0 >> 16) & 0xffffu)          // tensor_dim0[31:16]
        | (int)((dim1 & 0xffffu) << 16);         // tensor_dim1[15:0]
  g1[3] = (int)((dim1 >> 16) & 0xffffu)          // tensor_dim1[31:16]
        | (int)((dim0 & 0xffffu) << 16);         // tile_dim0 = dim0
  g1[4] = (int)(dim1 & 0xffffu);                 // tile_dim1 = dim1 (tile_dim2=0)
  g1[5] = (int)stride0;                          // tensor_dim0_stride[31:0]
  g1[6] = 0;                                     // stride hi + dim1_stride lo
  g1[7] = 0;
  v4i z4 = {0, 0, 0, 0};
#if defined(__clang_major__) && (__clang_major__ >= 23)
  v8i z8 = {0, 0, 0, 0, 0, 0, 0, 0};
  __builtin_amdgcn_tensor_load_to_lds(g0, g1, z4, z4, z8, 0);
#else
  __builtin_amdgcn_tensor_load_to_lds(g0, g1, z4, z4, 0);
#endif
}

// ---- workspace layout (float offsets) ----
#define OFF_X 0
#define OFF_Y (OFF_X + BATCH * PNZ)        // 20480
#define OFF_S (OFF_Y + BATCH * PNEQ)       // 25600
#define OFF_Z (OFF_S + BATCH * PNINEQ)     // 56320
#define OFF_D (OFF_Z + BATCH * PNINEQ)     // 87040
#define OFF_RC (OFF_D + BATCH * PNINEQ)    // 117760
#define OFF_RX (OFF_RC + BATCH * PNINEQ)   // 148480
#define OFF_RZ (OFF_RX + BATCH * PNZ)      // 168960
#define OFF_RY (OFF_RZ + BATCH * PNINEQ)   // 199680
#define OFF_RHS (OFF_RY + BATCH * PNEQ)    // 204800
#define OFF_T (OFF_RHS + BATCH * PNZ)      // 225280
#define OFF_GT (OFF_T + BATCH * PNINEQ)    // 256000
#define OFF_AT (OFF_GT + PNZ * PNINEQ)     // 294400
#define WS_FLOATS (OFF_AT + PNZ * ATPAD)   // 302080 floats = 1.18 MB

// ---- solve-kernel LDS layout (float offsets into dynamic smem) ----
#define SM_H 0                 // 160 x 161 (odd stride -> conflict-free columns)
#define SM_V 25760             // 160 x 41  ([A^T | u])
#define SM_SS 32320            // 40 x 41   (Schur [S | rhs])
#define SM_DD 33960            // 240       d
#define SM_DY 34200            // 40
#define SM_DX 34240            // 160
#define SM_T 34400             // 160
#define SM_DZ 34560            // 240
#define SM_DS 34800            // 240
#define SM_RED 35040           // 256
#define SM_SCAL 35296          // 4
#define SM_G 35300             // 240 x 160 TDM-staged copy of G
#define SM_FLOATS (SM_G + PNINEQ * PNZ)  // 73700 floats = 294800 B (< 320KB WGP LDS)

// ---------------------------------------------------------------------------
__global__ void pdipm_init(float* __restrict__ ws) {
  int idx = blockIdx.x * blockDim.x + threadIdx.x;
  // x,y = 0 ; s,z = 1
  if (idx < OFF_S) ws[idx] = 0.0f;
  else if (idx < OFF_D) ws[idx] = 1.0f;
}

__global__ void pdipm_trans(const float* __restrict__ G,
                            const float* __restrict__ A,
                            float* __restrict__ ws) {
  int idx = blockIdx.x * blockDim.x + threadIdx.x;
  if (idx < PNZ * PNINEQ) {
    int m = idx / PNINEQ, i = idx % PNINEQ;
    ws[OFF_GT + idx] = G[i * PNZ + m];
  } else if (idx < PNZ * PNINEQ + PNZ * ATPAD) {
    int j = idx - PNZ * PNINEQ;
    int m = j / ATPAD, e = j % ATPAD;
    ws[OFF_AT + j] = (e < PNEQ) ? A[e * PNZ + m] : 0.0f;
  }
}

// per-batch: mu, d, rcomp, and residual initializers
__global__ void pdipm_iter_setup(const float* __restrict__ qv,
                                 const float* __restrict__ hv,
                                 const float* __restrict__ bv,
                                 float* __restrict__ ws) {
  __shared__ float red[256];
  const int b = blockIdx.x, tid = threadIdx.x;
  const float* sv = ws + OFF_S + b * PNINEQ;
  const float* zv = ws + OFF_Z + b * PNINEQ;
  float acc = 0.0f;
  for (int i = tid; i < PNINEQ; i += 256) acc += sv[i] * zv[i];
  red[tid] = acc;
  __syncthreads();
  for (int o = 128; o > 0; o >>= 1) {
    if (tid < o) red[tid] += red[tid + o];
    __syncthreads();
  }
  const float mu = red[0] / (float)PNINEQ;
  float* dv = ws + OFF_D + b * PNINEQ;
  float* rc = ws + OFF_RC + b * PNINEQ;
  float* rz = ws + OFF_RZ + b * PNINEQ;
  for (int i = tid; i < PNINEQ; i += 256) {
    float si = sv[i], zi = zv[i];
    dv[i] = zi / si;
    rc[i] = si * zi - PSIGMA * mu;
    rz[i] = si - hv[b * PNINEQ + i];  // rz_ = x@G^T + s - h (GEMM adds x@G^T)
  }
  float* rx = ws + OFF_RX + b * PNZ;
  for (int n = tid; n < PNZ; n += 256) rx[n] = qv[b * PNZ + n];
  float* ry = ws + OFF_RY + b * PNEQ;
  if (tid < PNEQ) ry[tid] = -bv[b * PNEQ + tid];  // ry = x@A^T - b
}

// T = d*rz - rcomp/s ; rhs = -rx (then GEMM adds -(T@G))
__global__ void pdipm_tvec(float* __restrict__ ws) {
  const int b = blockIdx.x, tid = threadIdx.x;
  const float* dv = ws + OFF_D + b * PNINEQ;
  const float* rz = ws + OFF_RZ + b * PNINEQ;
  const float* rc = ws + OFF_RC + b * PNINEQ;
  const float* sv = ws + OFF_S + b * PNINEQ;
  float* tv = ws + OFF_T + b * PNINEQ;
  for (int i = tid; i < PNINEQ; i += 256)
    tv[i] = dv[i] * rz[i] - rc[i] / sv[i];
  const float* rx = ws + OFF_RX + b * PNZ;
  float* rhs = ws + OFF_RHS + b * PNZ;
  for (int n = tid; n < PNZ; n += 256) rhs[n] = -rx[n];
}

// OUT(128 x N) += alpha * X(128 x K) @ W(K x ldW), fp32 WMMA, one wave/block.
// All W buffers are padded so loads need no guards (only stores are guarded).
// A-lane: a[j] = X[bt*16 + lane%16][k + 2*(lane/16) + j]
// B-lane: b[j] = W[k + 2*(lane/16) + j][nt*16 + lane%16]
// C/D:    c[r] = OUT[bt*16 + r + 8*(lane/16)][col]
__global__ void wmma_gemm_acc(const float* __restrict__ X,
                              const float* __restrict__ W,
                              float* __restrict__ OUT,
                              int K, int N, int ldW, float alpha) {
  const int lane = threadIdx.x & 31;
  const int li = lane & 15;
  const int hh = lane >> 4;
  const int nt = blockIdx.x;
  const int bt = blockIdx.y;
  const int m = bt * 16 + li;
  const int nb = nt * 16 + li;
  v8f acc = {0.f, 0.f, 0.f, 0.f, 0.f, 0.f, 0.f, 0.f};
  for (int k = 0; k < K; k += 4) {
    const int k0 = k + 2 * hh;
    v2f av, bv;
    av.x = X[m * K + k0];
    av.y = X[m * K + k0 + 1];
    bv.x = W[k0 * ldW + nb];
    bv.y = W[(k0 + 1) * ldW + nb];
    acc = wmma4(av, bv, acc);
  }
  if (nb < N) {
    const int rowb = bt * 16 + 8 * hh;
#pragma unroll
    for (int r = 0; r < 8; ++r) OUT[(rowb + r) * N + nb] += alpha * acc[r];
  }
}

// ---------------------------------------------------------------------------
// One workgroup (256 thr = 8 waves) per batch element:
//   G staged to LDS by the Tensor Data Mover;
//   H = Q + G^T diag(d) G built in LDS via WMMA; Cholesky H = L L^T;
//   V = L^-1 [A^T | r1]; S = V^T V (Schur, SPD); S dy = V_A^T u + ry;
//   dx = L^-T (u - V_A dy); dz/ds; step size; in-place update of x,y,s,z.
// ---------------------------------------------------------------------------
__global__ void pdipm_solve(const float* __restrict__ Q,
                            const float* __restrict__ G,
                            const float* __restrict__ A,
                            float* __restrict__ ws) {
  extern __shared__ float sm[];
  float* Hs = sm + SM_H;    // stride 161
  float* Vs = sm + SM_V;    // stride 41
  float* Ss = sm + SM_SS;   // stride 41
  float* dds = sm + SM_DD;
  float* dys = sm + SM_DY;
  float* dxs = sm + SM_DX;
  float* tvs = sm + SM_T;
  float* dzs = sm + SM_DZ;
  float* dsv = sm + SM_DS;
  float* red = sm + SM_RED;
  float* scal = sm + SM_SCAL;
  float* Gs = sm + SM_G;    // 240 x 160 (row-major, as in global)

  const int b = blockIdx.x;
  const int tid = threadIdx.x;
  const int lane = tid & 31;
  const int wave = tid >> 5;
  const int li = lane & 15;
  const int hh = lane >> 4;

  // ---- Phase 0: TDM-stage G into LDS (wave 0 issues the DMA) + load d
  if (wave == 0) {
    tdm_load_2d_f32(G, (unsigned)(uintptr_t)(void*)Gs, PNZ, PNINEQ, PNZ);
    __builtin_amdgcn_s_wait_tensorcnt(0);
  }
  for (int i = tid; i < PNINEQ; i += 256) dds[i] = ws[OFF_D + b * PNINEQ + i];
  __syncthreads();

  // ---- Phase 1: H = Q + G^T diag(d) G  (WMMA f32, tiles striped over waves)
  for (int t = wave; t < 100; t += 8) {
    const int mt = t / 10, ntile = t % 10;
    const int mi = mt * 16 + li;
    const int ni = ntile * 16 + li;
    const int rowb = mt * 16 + 8 * hh;
    v8f acc;
#pragma unroll
    for (int r = 0; r < 8; ++r) acc[r] = Q[(rowb + r) * PNZ + ni];
    for (int k = 0; k < PNINEQ; k += 4) {
      const int k0 = k + 2 * hh;
      v2f av, bv;
      av.x = Gs[k0 * PNZ + mi];
      av.y = Gs[(k0 + 1) * PNZ + mi];
      bv.x = dds[k0] * Gs[k0 * PNZ + ni];
      bv.y = dds[k0 + 1] * Gs[(k0 + 1) * PNZ + ni];
      acc = wmma4(av, bv, acc);
    }
#pragma unroll
    for (int r = 0; r < 8; ++r) Hs[(rowb + r) * 161 + ni] = acc[r];
  }
  __syncthreads();

  // ---- Phase 2: Cholesky H = L L^T (lower, in place)
  for (int k = 0; k < PNZ; ++k) {
    if (tid == 0) scal[0] = sqrtf(Hs[k * 161 + k]);
    __syncthreads();
    const float lkk = scal[0];
    if (tid == 0) Hs[k * 161 + k] = lkk;
    for (int i = k + 1 + tid; i < PNZ; i += 256) Hs[i * 161 + k] /= lkk;
    __syncthreads();
    for (int i = k + 1 + tid; i < PNZ; i += 256) {
      const float lik = Hs[i * 161 + k];
      for (int j = k + 1; j <= i; ++j) Hs[i * 161 + j] -= lik * Hs[j * 161 + k];
    }
    __syncthreads();
  }

  // ---- Phase 3: V = L^-1 [A^T | r1]  (41 RHS columns)
  for (int idx = tid; idx < PNZ * PNEQ; idx += 256) {
    const int m = idx / PNEQ, e = idx % PNEQ;
    Vs[m * 41 + e] = A[e * PNZ + m];
  }
  for (int m = tid; m < PNZ; m += 256) Vs[m * 41 + 40] = ws[OFF_RHS + b * PNZ + m];
  __syncthreads();
  for (int k = 0; k < PNZ; ++k) {
    const float lkk = Hs[k * 161 + k];
    if (tid < 41) Vs[k * 41 + tid] /= lkk;
    __syncthreads();
    const int total = (PNZ - 1 - k) * 41;
    for (int idx = tid; idx < total; idx += 256) {
      const int i = k + 1 + idx / 41;
      const int c = idx % 41;
      Vs[i * 41 + c] -= Hs[i * 161 + k] * Vs[k * 41 + c];
    }
    __syncthreads();
  }

  // ---- Phase 4: Schur S = V_A^T V_A, rhs_y = V_A^T u + ry; solve S dy = rhs_y
  for (int idx = tid; idx < PNEQ * 41; idx += 256) {
    const int p = idx / 41, c = idx % 41;
    float a2 = 0.0f;
    for (int m = 0; m < PNZ; ++m) a2 += Vs[m * 41 + p] * Vs[m * 41 + c];
    Ss[idx] = a2;
  }
  __syncthreads();
  if (tid < PNEQ) Ss[tid * 41 + 40] += ws[OFF_RY + b * PNEQ + tid];
  __syncthreads();
  // Gaussian elimination (S is SPD; no pivoting needed)
  for (int k = 0; k < PNEQ; ++k) {
    const float piv = Ss[k * 41 + k];
    if (tid > k && tid <= 40) Ss[k * 41 + tid] /= piv;
    __syncthreads();
    const int cols = 40 - k;
    const int total = (PNEQ - 1 - k) * cols;
    for (int idx = tid; idx < total; idx += 256) {
      const int i = k + 1 + idx / cols;
      const int c = k + 1 + idx % cols;
      Ss[i * 41 + c] -= Ss[i * 41 + k] * Ss[k * 41 + c];
    }
    __syncthreads();
  }
  for (int k = PNEQ - 1; k >= 0; --k) {
    if (tid == 0) dys[k] = Ss[k * 41 + 40];
    __syncthreads();
    if (tid < k) Ss[tid * 41 + 40] -= Ss[tid * 41 + k] * dys[k];
    __syncthreads();
  }

  // ---- Phase 5: dx = L^-T (u - V_A dy)
  for (int m = tid; m < PNZ; m += 256) {
    float a2 = Vs[m * 41 + 40];
    for (int p = 0; p < PNEQ; ++p) a2 -= Vs[m * 41 + p] * dys[p];
    tvs[m] = a2;
  }
  __syncthreads();
  for (int k = PNZ - 1; k >= 0; --k) {
    if (tid == 0) dxs[k] = tvs[k] / Hs[k * 161 + k];
    __syncthreads();
    if (tid < k) tvs[tid] -= Hs[k * 161 + tid] * dxs[k];
    __syncthreads();
  }

  // ---- Phase 6: dz, ds, step size, updates (G read from LDS copy)
  const float* rz = ws + OFF_RZ + b * PNINEQ;
  const float* rc = ws + OFF_RC + b * PNINEQ;
  float* sv = ws + OFF_S + b * PNINEQ;
  float* zv = ws + OFF_Z + b * PNINEQ;
  for (int i = tid; i < PNINEQ; i += 256) {
    float gdx = 0.0f;
    for (int m = 0; m < PNZ; ++m) gdx += Gs[i * PNZ + m] * dxs[m];
    const float dz = dds[i] * (gdx + rz[i]) - rc[i] / sv[i];
    dzs[i] = dz;
    dsv[i] = -(rc[i] + sv[i] * dz) / zv[i];
  }
  __syncthreads();
  float loc = 1.0e9f;
  for (int i = tid; i < PNINEQ; i += 256) {
    const float dsi = dsv[i], dzi = dzs[i];
    if (dsi < 0.0f) loc = fminf(loc, -sv[i] / dsi);
    if (dzi < 0.0f) loc = fminf(loc, -zv[i] / dzi);
  }
  red[tid] = loc;
  __syncthreads();
  for (int o = 128; o > 0; o >>= 1) {
    if (tid < o) red[tid] = fminf(red[tid], red[tid + o]);
    __syncthreads();
  }
  const float alpha = fminf(1.0f, 0.99f * red[0]);
  float* xv = ws + OFF_X + b * PNZ;
  float* yv = ws + OFF_Y + b * PNEQ;
  for (int m = tid; m < PNZ; m += 256) xv[m] += alpha * dxs[m];
  if (tid < PNEQ) yv[tid] += alpha * dys[tid];
  for (int i = tid; i < PNINEQ; i += 256) {
    sv[i] += alpha * dsv[i];
    zv[i] += alpha * dzs[i];
  }
}

__global__ void pdipm_copy_out(const float* __restrict__ ws, float* __restrict__ out) {
  const int idx = blockIdx.x * blockDim.x + threadIdx.x;
  if (idx < BATCH * PNZ) out[idx] = ws[OFF_X + idx];
}

// ---------------------------------------------------------------------------
extern "C" void kernel_launch(void* const* d_in, const int* in_sizes, int n_in,
                              void* d_out, int out_size, void* d_ws, size_t ws_size,
                              hipStream_t stream) {
  (void)in_sizes; (void)n_in; (void)out_size; (void)ws_size;
  const float* Q = (const float*)d_in[0];   // 160x160 (symmetric)
  const float* qv = (const float*)d_in[1];  // 128x160
  const float* G = (const float*)d_in[2];   // 240x160
  const float* hv = (const float*)d_in[3];  // 128x240
  const float* A = (const float*)d_in[4];   // 40x160
  const float* bv = (const float*)d_in[5];  // 128x40
  float* ws = (float*)d_ws;
  float* out = (float*)d_out;

  pdipm_init<<<(OFF_D + 255) / 256, 256, 0, stream>>>(ws);
  pdipm_trans<<<(PNZ * PNINEQ + PNZ * ATPAD + 255) / 256, 256, 0, stream>>>(G, A, ws);

  const size_t smem = (size_t)SM_FLOATS * sizeof(float);
  for (int it = 0; it < NITER; ++it) {
    pdipm_iter_setup<<<BATCH, 256, 0, stream>>>(qv, hv, bv, ws);
    // rx = q + x@Q + y@A + z@G
    wmma_gemm_acc<<<dim3(10, 8), 32, 0, stream>>>(ws + OFF_X, Q, ws + OFF_RX, PNZ, PNZ, PNZ, 1.0f);
    wmma_gemm_acc<<<dim3(10, 8), 32, 0, stream>>>(ws + OFF_Y, A, ws + OFF_RX, PNEQ, PNZ, PNZ, 1.0f);
    wmma_gemm_acc<<<dim3(10, 8), 32, 0, stream>>>(ws + OFF_Z, G, ws + OFF_RX, PNINEQ, PNZ, PNZ, 1.0f);
    // rz_ = (s - h) + x@G^T ; ry = -b + x@A^T (A^T padded to stride 48)
    wmma_gemm_acc<<<dim3(15, 8), 32, 0, stream>>>(ws + OFF_X, ws + OFF_GT, ws + OFF_RZ, PNZ, PNINEQ, PNINEQ, 1.0f);
    wmma_gemm_acc<<<dim3(3, 8), 32, 0, stream>>>(ws + OFF_X, ws + OFF_AT, ws + OFF_RY, PNZ, PNEQ, ATPAD, 1.0f);
    // rhs_x = -(rx + T@G)
    pdipm_tvec<<<BATCH, 256, 0, stream>>>(ws);
    wmma_gemm_acc<<<dim3(10, 8), 32, 0, stream>>>(ws + OFF_T, G, ws + OFF_RHS, PNINEQ, PNZ, PNZ, -1.0f);
    // KKT solve + update
    pdipm_solve<<<BATCH, 256, smem, stream>>>(Q, G, A, ws);
  }
  pdipm_copy_out<<<(BATCH * PNZ + 255) / 256, 256, 0, stream>>>(ws, out);
}